// GAT_47682726920882
// MI455X (gfx1250) — compile-verified
//
#include <hip/hip_runtime.h>
#include <hip/hip_bf16.h>

typedef __attribute__((ext_vector_type(2))) float v2f;
typedef __attribute__((ext_vector_type(8))) float v8f;

#define NEG_SLOPE 0.2f

__device__ __forceinline__ float lrelu(float x) { return x > 0.f ? x : NEG_SLOPE * x; }

// Order-preserving float<->uint encoding for atomic max over signed floats.
__device__ __forceinline__ unsigned encf(float f) {
  unsigned b = __float_as_uint(f);
  return (b & 0x80000000u) ? ~b : (b | 0x80000000u);
}
__device__ __forceinline__ float decf(unsigned e) {
  unsigned b = (e & 0x80000000u) ? (e & 0x7FFFFFFFu) : ~e;
  return __uint_as_float(b);
}
// encf(-inf) == 0x007FFFFF
#define ENC_NEG_INF 0x007FFFFFu

// ---------------------------------------------------------------------------
// Per-layer state init: deg=1 (self loop), den=0, running max = -inf
// ---------------------------------------------------------------------------
__global__ void init_layer(int* __restrict__ deg, float* __restrict__ den,
                           unsigned* __restrict__ mEnc, int n, int H) {
  int i = blockIdx.x * blockDim.x + threadIdx.x;
  if (i >= n) return;
  deg[i] = 1;
  for (int h = 0; h < H; ++h) {
    den[i * H + h] = 0.f;
    mEnc[i * H + h] = ENC_NEG_INF;
  }
}

// ---------------------------------------------------------------------------
// Dual GEMM: xl = x@Wl + bl, xr = x@Wr + br  using V_WMMA_F32_16X16X4_F32.
// Block = 256 threads = 8 waves. Wave w: matrix = w/4 (l/r), coltile = w%4.
// Each wave produces a 16x16 fp32 tile; K-loop steps 4 per WMMA.
// A frag (16x4 f32): lane m=lane&15 is row M; VGPR0/1 = K={0,1} (lanes 0-15)
//                    or K={2,3} (lanes 16-31).   B frag mirrors with N=lane&15.
// ---------------------------------------------------------------------------
__global__ void gemm_xlxr(const float* __restrict__ x, int ldx,
                          const float* __restrict__ Wl, const float* __restrict__ bl,
                          const float* __restrict__ Wr, const float* __restrict__ br,
                          float* __restrict__ xl, float* __restrict__ xr, int n) {
  const int wave = threadIdx.x >> 5;
  const int lane = threadIdx.x & 31;
  const int mat = wave >> 2;            // 0 -> (Wl,bl,xl), 1 -> (Wr,br,xr)
  const int colbase = (wave & 3) * 16;  // 64 output cols -> 4 tiles
  const int rowbase = blockIdx.x * 16;

  const float* W = mat ? Wr : Wl;
  const float* bias = mat ? br : bl;
  float* out = mat ? xr : xl;

  const int m = lane & 15;
  const int koff = (lane < 16) ? 0 : 2;

  int rr = rowbase + m;
  if (rr >= n) rr = n - 1;  // clamped read row (stores are guarded)
  const float* xrow = x + (size_t)rr * ldx;

  v8f acc = {};
  for (int k = 0; k < ldx; k += 4) {
    v2f a, b;
    a.x = xrow[k + koff];
    a.y = xrow[k + koff + 1];
    b.x = W[(size_t)(k + koff) * 64 + colbase + m];
    b.y = W[(size_t)(k + koff + 1) * 64 + colbase + m];
    acc = __builtin_amdgcn_wmma_f32_16x16x4_f32(
        /*neg_a=*/false, a, /*neg_b=*/false, b,
        /*c_mod=*/(short)0, acc, /*reuse_a=*/false, /*reuse_b=*/false);
  }

  const float bb = bias[colbase + m];
  const int rsel = (lane < 16) ? 0 : 8;  // C/D layout: VGPR r -> M=r (lo) / r+8 (hi)
#pragma unroll
  for (int r = 0; r < 8; ++r) {
    int row = rowbase + r + rsel;
    if (row < n) out[(size_t)row * 64 + colbase + m] = acc[r] + bb;
  }
}

// ---------------------------------------------------------------------------
// CSR build: histogram -> 2-level exclusive scan -> scatter
// ---------------------------------------------------------------------------
__global__ void count_deg(const int* __restrict__ ei, int E, int* __restrict__ deg) {
  int e = blockIdx.x * blockDim.x + threadIdx.x;
  if (e < E) atomicAdd(&deg[ei[E + e]], 1);
}

__global__ void scan_blk(const int* __restrict__ deg, int* __restrict__ excl,
                         int* __restrict__ blockSums, int n) {
  __shared__ int sh[256];
  const int t = threadIdx.x;
  const int base = blockIdx.x * 1024 + t * 4;
  int v0 = 0, v1 = 0, v2 = 0, v3 = 0;
  if (base + 0 < n) v0 = deg[base + 0];
  if (base + 1 < n) v1 = deg[base + 1];
  if (base + 2 < n) v2 = deg[base + 2];
  if (base + 3 < n) v3 = deg[base + 3];
  const int s = v0 + v1 + v2 + v3;
  sh[t] = s;
  __syncthreads();
  for (int off = 1; off < 256; off <<= 1) {
    int add = (t >= off) ? sh[t - off] : 0;
    __syncthreads();
    sh[t] += add;
    __syncthreads();
  }
  const int ex = sh[t] - s;
  if (base + 0 < n) excl[base + 0] = ex;
  if (base + 1 < n) excl[base + 1] = ex + v0;
  if (base + 2 < n) excl[base + 2] = ex + v0 + v1;
  if (base + 3 < n) excl[base + 3] = ex + v0 + v1 + v2;
  if (t == 255) blockSums[blockIdx.x] = sh[255];
}

__global__ void scan_top(const int* __restrict__ blockSums,
                         int* __restrict__ blockPrefix, int nb) {
  __shared__ int sh[1024];
  const int t = threadIdx.x;
  const int v = (t < nb) ? blockSums[t] : 0;
  sh[t] = v;
  __syncthreads();
  for (int off = 1; off < 1024; off <<= 1) {
    int add = (t >= off) ? sh[t - off] : 0;
    __syncthreads();
    sh[t] += add;
    __syncthreads();
  }
  if (t < nb) blockPrefix[t] = sh[t] - v;   // exclusive
  if (t == 0) blockPrefix[nb] = sh[1023];   // grand total
}

__global__ void scan_add(const int* __restrict__ excl, const int* __restrict__ blockPrefix,
                         int* __restrict__ offsets, int* __restrict__ cursor, int n, int nb) {
  int i = blockIdx.x * blockDim.x + threadIdx.x;
  if (i < n) {
    int off = excl[i] + blockPrefix[i >> 10];
    offsets[i] = off;
    cursor[i] = off;
  }
  if (i == 0) offsets[n] = blockPrefix[nb];
}

__global__ void scatter_edges(const int* __restrict__ ei, int E, int n,
                              int* __restrict__ cursor, int* __restrict__ csr_src,
                              int* __restrict__ csr_eid) {
  int t = blockIdx.x * blockDim.x + threadIdx.x;
  if (t >= E + n) return;
  int s, d;
  if (t < E) { s = ei[t]; d = ei[E + t]; } else { s = t - E; d = s; }
  int pos = atomicAdd(&cursor[d], 1);
  csr_src[pos] = s;
  csr_eid[pos] = t;
}

// ---------------------------------------------------------------------------
// Pass 1: per-(edge,head) attention logit + running segment max (encoded).
// ---------------------------------------------------------------------------
__global__ void edge_logits(const int* __restrict__ ei, int E, int n,
                            const float* __restrict__ xl, const float* __restrict__ xr,
                            const float* __restrict__ att, float* __restrict__ aBuf,
                            unsigned* __restrict__ mEnc, int H, int C) {
  const int t = blockIdx.x * blockDim.x + threadIdx.x;
  const int total = (E + n) * H;
  if (t >= total) return;
  const int e = t / H, h = t - e * H;
  int s, d;
  if (e < E) { s = ei[e]; d = ei[E + e]; } else { s = e - E; d = s; }
  const float4* L = (const float4*)(xl + (size_t)s * 64 + h * C);
  const float4* R = (const float4*)(xr + (size_t)d * 64 + h * C);
  const float4* A = (const float4*)(att + h * C);
  float a = 0.f;
  const int q = C >> 2;
  for (int c = 0; c < q; ++c) {
    float4 l = L[c], r = R[c], w = A[c];
    a += lrelu(l.x + r.x) * w.x + lrelu(l.y + r.y) * w.y +
         lrelu(l.z + r.z) * w.z + lrelu(l.w + r.w) * w.w;
  }
  aBuf[t] = a;
  atomicMax(&mEnc[d * H + h], encf(a));
}

// ---------------------------------------------------------------------------
// Pass 2: w = exp(a - max), accumulate softmax denominators.
// ---------------------------------------------------------------------------
__global__ void edge_expsum(const int* __restrict__ ei, int E, int n,
                            float* __restrict__ aBuf, const unsigned* __restrict__ mEnc,
                            float* __restrict__ den, int H) {
  const int t = blockIdx.x * blockDim.x + threadIdx.x;
  const int total = (E + n) * H;
  if (t >= total) return;
  const int e = t / H, h = t - e * H;
  const int d = (e < E) ? ei[E + e] : (e - E);
  const float m = decf(mEnc[d * H + h]);
  const float w = __expf(aBuf[t] - m);
  aBuf[t] = w;
  atomicAdd(&den[d * H + h], w);
}

// ---------------------------------------------------------------------------
// Pass 3: per-destination aggregation (one wave32 per node, 2 chans/lane) with
// fused bias + (relu) + L1-norm; final layer additionally L2-norm + relu.
// ---------------------------------------------------------------------------
__global__ void aggregate(const int* __restrict__ offsets, const int* __restrict__ csr_src,
                          const int* __restrict__ csr_eid, const float* __restrict__ xl,
                          const float* __restrict__ wBuf, const float* __restrict__ den,
                          const float* __restrict__ bias, float* __restrict__ out,
                          int n, int H, int C, int mode) {
  const int wave = threadIdx.x >> 5;
  const int lane = threadIdx.x & 31;
  const int node = blockIdx.x * 8 + wave;
  if (node >= n) return;

  const int j1 = lane, j2 = lane + 32;
  const int h1 = j1 / C, h2 = j2 / C;
  const float inv1 = 1.f / (den[node * H + h1] + 1e-16f);
  const float inv2 = 1.f / (den[node * H + h2] + 1e-16f);

  float acc1 = 0.f, acc2 = 0.f;
  const int s0 = offsets[node], s1 = offsets[node + 1];
  for (int s = s0; s < s1; ++s) {
    const int sr = csr_src[s];
    const int eid = csr_eid[s];
    const float al1 = wBuf[eid * H + h1] * inv1;
    const float al2 = wBuf[eid * H + h2] * inv2;
    acc1 += xl[(size_t)sr * 64 + j1] * al1;
    acc2 += xl[(size_t)sr * 64 + j2] * al2;
  }

  float v1 = acc1 + bias[j1];
  float v2 = acc2 + bias[j2];
  if (mode == 0) { v1 = fmaxf(v1, 0.f); v2 = fmaxf(v2, 0.f); }  // relu (non-final)

  // L1 normalize across the wave's 64 channels
  float s = fabsf(v1) + fabsf(v2);
  for (int m = 16; m >= 1; m >>= 1) s += __shfl_xor(s, m, 32);
  const float invl1 = 1.f / fmaxf(s, 1e-12f);
  v1 *= invl1;
  v2 *= invl1;

  if (mode == 1) {  // final: L2 normalize + relu
    float q = v1 * v1 + v2 * v2;
    for (int m = 16; m >= 1; m >>= 1) q += __shfl_xor(q, m, 32);
    const float invl2 = 1.f / fmaxf(sqrtf(q), 1e-12f);
    v1 = fmaxf(v1 * invl2, 0.f);
    v2 = fmaxf(v2 * invl2, 0.f);
  }

  out[(size_t)node * 64 + j1] = v1;
  out[(size_t)node * 64 + j2] = v2;
}

// ---------------------------------------------------------------------------
// Orchestration
// ---------------------------------------------------------------------------
extern "C" void kernel_launch(void* const* d_in, const int* in_sizes, int n_in,
                              void* d_out, int out_size, void* d_ws, size_t ws_size,
                              hipStream_t stream) {
  const int N = in_sizes[0] / 128;
  const int E = in_sizes[1] / 2;

  const float* x0 = (const float*)d_in[0];
  const int* ei[3] = {(const int*)d_in[1], (const int*)d_in[2], (const int*)d_in[3]};

  char* p = (char*)d_ws;
  auto alloc = [&](size_t bytes) -> void* {
    void* r = (void*)p;
    p += (bytes + 255) & ~(size_t)255;
    return r;
  };
  float* xl = (float*)alloc((size_t)N * 64 * 4);
  float* xr = (float*)alloc((size_t)N * 64 * 4);
  float* xbuf = (float*)alloc((size_t)N * 64 * 4);
  float* aBuf = (float*)alloc((size_t)(E + N) * 4 * 4);  // logits then exp-weights
  float* den = (float*)alloc((size_t)N * 4 * 4);
  unsigned* mEnc = (unsigned*)alloc((size_t)N * 4 * 4);
  int* deg = (int*)alloc((size_t)N * 4);
  int* excl = (int*)alloc((size_t)N * 4);
  const int nb = (N + 1023) / 1024;
  int* blockSums = (int*)alloc((size_t)(nb + 1) * 4);
  int* blockPrefix = (int*)alloc((size_t)(nb + 1) * 4);
  int* offsets = (int*)alloc((size_t)(N + 1) * 4);
  int* cursor = (int*)alloc((size_t)N * 4);
  int* csr_src = (int*)alloc((size_t)(E + N) * 4);
  int* csr_eid = (int*)alloc((size_t)(E + N) * 4);

  struct Cfg { const float* x; int ldx; int H; int C; int wi; float* out; int mode; };
  const Cfg cfg[3] = {
      {x0, 128, 4, 16, 4, xbuf, 0},
      {xbuf, 64, 4, 16, 10, xbuf, 0},
      {xbuf, 64, 1, 64, 16, (float*)d_out, 1},
  };

  for (int L = 0; L < 3; ++L) {
    const Cfg& c = cfg[L];
    const float* Wl = (const float*)d_in[c.wi + 0];
    const float* bl = (const float*)d_in[c.wi + 1];
    const float* Wr = (const float*)d_in[c.wi + 2];
    const float* br = (const float*)d_in[c.wi + 3];
    const float* att = (const float*)d_in[c.wi + 4];
    const float* bias = (const float*)d_in[c.wi + 5];

    init_layer<<<dim3((N + 255) / 256), dim3(256), 0, stream>>>(deg, den, mEnc, N, c.H);
    gemm_xlxr<<<dim3((N + 15) / 16), dim3(256), 0, stream>>>(c.x, c.ldx, Wl, bl, Wr, br, xl, xr, N);
    count_deg<<<dim3((E + 255) / 256), dim3(256), 0, stream>>>(ei[L], E, deg);
    scan_blk<<<dim3(nb), dim3(256), 0, stream>>>(deg, excl, blockSums, N);
    scan_top<<<dim3(1), dim3(1024), 0, stream>>>(blockSums, blockPrefix, nb);
    scan_add<<<dim3((N + 255) / 256), dim3(256), 0, stream>>>(excl, blockPrefix, offsets, cursor, N, nb);
    scatter_edges<<<dim3((E + N + 255) / 256), dim3(256), 0, stream>>>(ei[L], E, N, cursor, csr_src, csr_eid);

    const int totEH = (E + N) * c.H;
    edge_logits<<<dim3((totEH + 255) / 256), dim3(256), 0, stream>>>(ei[L], E, N, xl, xr, att, aBuf, mEnc, c.H, c.C);
    edge_expsum<<<dim3((totEH + 255) / 256), dim3(256), 0, stream>>>(ei[L], E, N, aBuf, mEnc, den, c.H);
    aggregate<<<dim3((N + 7) / 8), dim3(256), 0, stream>>>(offsets, csr_src, csr_eid, xl, aBuf, den, bias, c.out, N, c.H, c.C, c.mode);
  }
}